// SegmentedMean_87454124082187
// MI455X (gfx1250) — compile-verified
//
#include <hip/hip_runtime.h>

typedef __attribute__((ext_vector_type(2))) float v2f;
typedef __attribute__((ext_vector_type(8))) float v8f;

#define COLS 64
#define RPB  128  // rows per block (8 windows)
#define SW   16   // window rows: max 16 distinct segments -> 16 WMMA slots always suffice

// ---------------------------------------------------------------------------
// Kernel 1: zero sums (d_out) and counts (d_ws)
// ---------------------------------------------------------------------------
__global__ void seg_init(float* __restrict__ out, float* __restrict__ counts,
                         int out_n, int seg_n) {
  int i = blockIdx.x * blockDim.x + threadIdx.x;
  int stride = gridDim.x * blockDim.x;
  for (int j = i; j < out_n; j += stride) out[j] = 0.0f;
  for (int j = i; j < seg_n; j += stride) counts[j] = 0.0f;
}

// ---------------------------------------------------------------------------
// Kernel 2: segmented sum via one-hot WMMA (f32 16x16x4).
//   Grid covers only full RPB blocks: no bounds checks inside.
//   block = 128 threads = 4 waves; wave w owns columns [16w, 16w+16).
// ---------------------------------------------------------------------------
__global__ void __launch_bounds__(128)
seg_accum(const float* __restrict__ F, const int* __restrict__ S,
          float* __restrict__ out, float* __restrict__ counts) {
  const int lane = threadIdx.x & 31;
  const int wave = threadIdx.x >> 5;   // column group 0..3
  const int n    = lane & 15;          // column within group (B/D layout)
  const int g    = lane >> 4;          // lane half (K / slot-half select)
  const int col0 = wave * 16;
  const int rowbase = blockIdx.x * RPB;

  const int  myslot    = lane & 15;                  // slot owned by this lane
  const bool in_lo     = (lane >= 1) && (lane <= 15);
  const int  posn_addr = (myslot + 1) << 2;          // bpermute byte addr of slot+1

  for (int sw = 0; sw < RPB; sw += SW) {
    const int rb = rowbase + sw;

    // ---- window metadata: boundaries -> ranks -> slot intervals ----
    const int  seg     = S[rb + myslot];             // halves load duplicates (benign)
    const int  segprev = __shfl_up(seg, 1);
    const bool bnd     = in_lo && (seg != segprev);
    const unsigned bmask = (unsigned)__ballot(bnd) & 0xFFFEu;
    const int  maxrank = __popc(bmask);              // uniform: last used slot
    const int  myrank  = __popc(bmask & ((2u << myslot) - 1u));  // rank of row 'myslot'

    // slot -> segment id: scatter (duplicate writers carry identical values)
    const int slot_seg = __builtin_amdgcn_ds_permute(myrank << 2, seg);
    // slot -> first row: unique writer per slot (run starts); others -> trash lanes 16..31
    const bool run_start = (lane == 0) || bnd;
    const int  dstl      = run_start ? myrank : (16 + myslot);
    const int  pos_raw   = __builtin_amdgcn_ds_permute(dstl << 2, myslot);
    const int  posn_raw  = __builtin_amdgcn_ds_bpermute(posn_addr, pos_raw);

    const bool vslot = (myslot <= maxrank);
    const int  pos   = vslot ? pos_raw : SW;                          // empty if unused
    const int  posn  = (myslot == maxrank) ? SW : (vslot ? posn_raw : SW);

    // ---- prefetch next window's rows ----
    if (sw + SW < RPB)
      __builtin_prefetch(&F[(size_t)(rb + SW + myslot) * COLS + col0 + n], 0, 0);

    // ---- WMMA: D[slot][col] += onehot x F, K=4 per step, 4 steps ----
    v8f acc = {0.0f, 0.0f, 0.0f, 0.0f, 0.0f, 0.0f, 0.0f, 0.0f};
#pragma unroll
    for (int k0 = 0; k0 < SW; k0 += 4) {
      const int kx = k0 + 2 * g;       // f32 A/B 16x4 layout: v0 = K0|K2, v1 = K1|K3
      const int ky = kx + 1;
      v2f a, b;
      a.x = (kx >= pos && kx < posn) ? 1.0f : 0.0f;   // row kx belongs to slot 'myslot'?
      a.y = (ky >= pos && ky < posn) ? 1.0f : 0.0f;
      b.x = F[(size_t)(rb + kx) * COLS + col0 + n];   // two 64B lines per wave: coalesced
      b.y = F[(size_t)(rb + ky) * COLS + col0 + n];
      acc = __builtin_amdgcn_wmma_f32_16x16x4_f32(false, a, false, b,
                                                  (short)0, acc, false, false);
    }

    // ---- flush used slots: avg ~1.8 iterations (uniform early exit) ----
#pragma unroll
    for (int r = 0; r < 8; ++r) {
      if (r > maxrank) break;                       // uniform; exact for both halves
      const int m   = r + 8 * g;                    // D row held in acc[r] by this half
      const int sid = __shfl(slot_seg, m);          // all lanes active for bpermute
      if (m <= maxrank)
        atomicAdd(&out[(size_t)sid * COLS + col0 + n], acc[r]);
    }

    // ---- counts: one wave only; count = interval length ----
    if (wave == 0) {
      if (lane < 16 && vslot)
        atomicAdd(&counts[slot_seg], (float)(posn - pos));
    }
  }
}

// ---------------------------------------------------------------------------
// Kernel 2b: scalar tail for rows not covered by full blocks (usually none)
// ---------------------------------------------------------------------------
__global__ void seg_tail(const float* __restrict__ F, const int* __restrict__ S,
                         float* __restrict__ out, float* __restrict__ counts,
                         int start, int nrows) {
  const int c = threadIdx.x;  // 64 threads, one per column
  for (int r = start; r < nrows; ++r) {
    const int s = S[r];
    atomicAdd(&out[(size_t)s * COLS + c], F[(size_t)r * COLS + c]);
    if (c == 0) atomicAdd(&counts[s], 1.0f);
  }
}

// ---------------------------------------------------------------------------
// Kernel 3: mean = sum / count; empty segments -> 0
// ---------------------------------------------------------------------------
__global__ void seg_final(float* __restrict__ out, const float* __restrict__ counts,
                          int out_n) {
  int i = blockIdx.x * blockDim.x + threadIdx.x;
  int stride = gridDim.x * blockDim.x;
  for (int j = i; j < out_n; j += stride) {
    float c = counts[j >> 6];   // COLS == 64
    float s = out[j];
    out[j] = (c > 0.0f) ? (s / c) : 0.0f;
  }
}

// ---------------------------------------------------------------------------
extern "C" void kernel_launch(void* const* d_in, const int* in_sizes, int n_in,
                              void* d_out, int out_size, void* d_ws, size_t ws_size,
                              hipStream_t stream) {
  const float* F = (const float*)d_in[0];   // [nrows, 64] f32
  const int*   S = (const int*)d_in[1];     // [nrows] segment ids (sorted)
  float* out    = (float*)d_out;            // [nseg, 64] f32
  float* counts = (float*)d_ws;             // [nseg] f32 scratch

  const int nrows  = in_sizes[1];
  const int nseg   = out_size / COLS;
  const int nblk   = nrows / RPB;           // full blocks only (2M/128 = 15625 exact)
  const int tail0  = nblk * RPB;

  seg_init<<<2048, 256, 0, stream>>>(out, counts, out_size, nseg);

  if (nblk > 0)
    seg_accum<<<nblk, 128, 0, stream>>>(F, S, out, counts);
  if (tail0 < nrows)
    seg_tail<<<1, COLS, 0, stream>>>(F, S, out, counts, tail0, nrows);

  seg_final<<<2048, 256, 0, stream>>>(out, counts, out_size);
}